// GraphConvLayer_25245817766095
// MI455X (gfx1250) — compile-verified
//
#include <hip/hip_runtime.h>
#include <hip/hip_bf16.h>

// GCN layer for MI455X (gfx1250, wave32):
//   deg = segment_sum(ones, row);  dis = rsqrt(deg) (0 if deg==0)
//   prop[i] = sum_{e: row_e==i} dis[row_e]*dis[col_e] * feat[col_e]
//   out = relu(prop @ W + bias)
// Dense stage uses V_WMMA_F32_16X16X4_F32 (full f32 precision, matches ref).

typedef float v2f __attribute__((ext_vector_type(2)));
typedef float v8f __attribute__((ext_vector_type(8)));

__global__ void zero_ws_kernel(float* __restrict__ p, size_t n) {
    size_t i = (size_t)blockIdx.x * blockDim.x + threadIdx.x;
    size_t stride = (size_t)gridDim.x * blockDim.x;
    for (; i < n; i += stride) p[i] = 0.0f;
}

__global__ void degree_kernel(const long long* __restrict__ ei,
                              float* __restrict__ deg, long long E) {
    long long e = (long long)blockIdx.x * blockDim.x + threadIdx.x;
    if (e < E) {
        long long r = ei[e];  // edge_index[0, e]
        atomicAdd(&deg[r], 1.0f);
    }
}

__global__ void rsqrt_kernel(float* __restrict__ deg, int n) {
    int i = blockIdx.x * blockDim.x + threadIdx.x;
    if (i < n) {
        float d = deg[i];
        deg[i] = (d > 0.0f) ? rsqrtf(d) : 0.0f;  // deg==0 -> inf -> 0 per ref
    }
}

// One wave32 per edge; lane l handles feature elements 2l, 2l+1 (float2 = 8B
// per lane -> fully coalesced 256B row reads; feature table is L2-resident).
__global__ void spmm_kernel(const long long* __restrict__ ei,
                            const float* __restrict__ feat,
                            const float* __restrict__ dis,
                            float* __restrict__ prop, long long E) {
    long long t = (long long)blockIdx.x * blockDim.x + threadIdx.x;
    long long e = t >> 5;
    int lane = threadIdx.x & 31;
    if (e >= E) return;
    long long r = ei[e];
    long long c = ei[E + e];
    float w = dis[r] * dis[c];
    const float2* __restrict__ src = (const float2*)(feat + c * 64);
    float2 v = src[lane];
    float* dst = prop + r * 64 + 2 * lane;
    atomicAdd(dst + 0, v.x * w);
    atomicAdd(dst + 1, v.y * w);
}

// Dense stage: out = relu(prop @ W + bias).
// 256-thread block = 8 waves; wave w: rowTile = 2*blockIdx.x + (w>>2),
// colTile = w&3. Each wave builds one 16x16 f32 tile via 16 chained
// V_WMMA_F32_16X16X4_F32 ops over K=64.
__global__ void gemm_bias_relu_kernel(const float* __restrict__ A,   // [N,64]
                                      const float* __restrict__ W,   // [64,64]
                                      const float* __restrict__ bias,// [64]
                                      float* __restrict__ out,       // [N,64]
                                      int N) {
    const int lane = threadIdx.x & 31;
    const int wave = threadIdx.x >> 5;            // 0..7
    const int rowBase = (blockIdx.x * 2 + (wave >> 2)) * 16;
    const int colBase = (wave & 3) * 16;
    if (rowBase >= N) return;                     // wave-uniform: EXEC stays full

    const int half = lane >> 4;                   // 0 or 1
    const int l15  = lane & 15;

    // Clamp row for loads so EXEC remains all-1s through the WMMA chain.
    int ar = rowBase + l15;
    if (ar >= N) ar = N - 1;
    const float* __restrict__ arow = A + (size_t)ar * 64;

#if __has_builtin(__builtin_amdgcn_wmma_f32_16x16x4_f32)
    v8f acc = {};
#pragma unroll
    for (int k0 = 0; k0 < 64; k0 += 4) {
        int kk = k0 + 2 * half;
        // A frag (16x4): lane -> A[row=l15][kk], A[row][kk+1]
        v2f a;
        a.x = arow[kk];
        a.y = arow[kk + 1];
        // B frag (4x16): lane -> W[kk][colBase+l15], W[kk+1][colBase+l15]
        v2f b;
        b.x = W[(size_t)kk * 64 + colBase + l15];
        b.y = W[(size_t)(kk + 1) * 64 + colBase + l15];
        acc = __builtin_amdgcn_wmma_f32_16x16x4_f32(
            /*neg_a=*/false, a, /*neg_b=*/false, b,
            /*c_mod=*/(short)0, acc, /*reuse_a=*/false, /*reuse_b=*/false);
    }
    float bv = bias[colBase + l15];
#pragma unroll
    for (int r = 0; r < 8; ++r) {
        int m = rowBase + r + 8 * half;           // C/D layout: M = r + 8*half
        if (m < N) {
            float v = acc[r] + bv;
            out[(size_t)m * 64 + colBase + l15] = v > 0.0f ? v : 0.0f;
        }
    }
#else
    // Scalar fallback (should not be taken on gfx1250 toolchains).
    float accs[8] = {};
#pragma unroll
    for (int k = 0; k < 64; ++k) {
        float bvk = W[(size_t)k * 64 + colBase + l15];
#pragma unroll
        for (int r = 0; r < 8; ++r) {
            int m = rowBase + r + 8 * half;
            int mm = m >= N ? N - 1 : m;
            accs[r] = fmaf(A[(size_t)mm * 64 + k], bvk, accs[r]);
        }
    }
    float bv = bias[colBase + l15];
#pragma unroll
    for (int r = 0; r < 8; ++r) {
        int m = rowBase + r + 8 * half;
        if (m < N) {
            float v = accs[r] + bv;
            out[(size_t)m * 64 + colBase + l15] = v > 0.0f ? v : 0.0f;
        }
    }
#endif
}

extern "C" void kernel_launch(void* const* d_in, const int* in_sizes, int n_in,
                              void* d_out, int out_size, void* d_ws, size_t ws_size,
                              hipStream_t stream) {
    const float*     feat = (const float*)d_in[0];      // [N,64] f32
    const long long* ei   = (const long long*)d_in[1];  // [2,E] int64
    const float*     W    = (const float*)d_in[2];      // [64,64] f32
    const float*     bias = (const float*)d_in[3];      // [64] f32
    float*           out  = (float*)d_out;              // [N,64] f32

    const int       N = in_sizes[0] / 64;
    const long long E = (long long)in_sizes[1] / 2;

    float* deg  = (float*)d_ws;          // [N]
    float* prop = deg + N;               // [N,64]

    // 1) zero scratch (deg + prop contiguously)
    {
        size_t total = (size_t)N * 65;
        int blocks = 2048;
        zero_ws_kernel<<<blocks, 256, 0, stream>>>(deg, total);
    }
    // 2) degrees
    {
        long long blocks = (E + 255) / 256;
        degree_kernel<<<(int)blocks, 256, 0, stream>>>(ei, deg, E);
    }
    // 3) deg^-1/2 in place
    {
        int blocks = (N + 255) / 256;
        rsqrt_kernel<<<blocks, 256, 0, stream>>>(deg, N);
    }
    // 4) normalized scatter-add SpMM (one wave per edge)
    {
        long long threads = E * 32;
        long long blocks = (threads + 255) / 256;
        spmm_kernel<<<(int)blocks, 256, 0, stream>>>(ei, feat, deg, prop, E);
    }
    // 5) dense GEMM + bias + relu via WMMA f32 16x16x4
    {
        int blocks = (N + 31) / 32;
        gemm_bias_relu_kernel<<<blocks, 256, 0, stream>>>(prop, W, bias, out, N);
    }
}